// YOLOLoss_89283780149525
// MI455X (gfx1250) — compile-verified
//
#include <hip/hip_runtime.h>

// ---------------------------------------------------------------------------
// YOLO loss for gfx1250 (MI455X).  Memory-streaming reduction:
//   - noobj pass dominates: 409,600 strided f32 reads (~52 MB of cachelines)
//     -> ~2.3 us roofline at 23.3 TB/s HBM.  Streamed contiguously per block.
//   - WMMA f32 16x16x4 (A = ones) used as an accumulator on the matrix pipe:
//     sum(D - C) == 16 * sum(B) independent of the exact lane layout, since
//     A/B VGPR layouts are bijections onto the 64 matrix slots.  Co-executes
//     with the exp/log1p transcendentals of softplus.
//   - All reductions are deterministic (fixed-order wave/LDS/serial combine;
//     only integer atomicOr, which is exactly commutative).
// ---------------------------------------------------------------------------

typedef __attribute__((ext_vector_type(2))) float v2f;
typedef __attribute__((ext_vector_type(8))) float v8f;

#define LAMBDA_COORD 5.0f
#define LAMBDA_NOOBJ 0.5f

constexpr int B     = 64;
constexpr int G     = 80;
constexpr int NC    = 80;
constexpr int C5    = NC + 5;          // 85
constexpr int T     = 50;
constexpr int NT    = B * T;           // 3200 targets
constexpr int CELLS = B * G * G;       // 409600 cells
constexpr int OCCW  = CELLS / 32;      // 12800 mask words

#define TPB  256
#define NBLK 100
#define PER  (CELLS / (TPB * NBLK))    // 16 cells per thread (uniform, no tail)

#if defined(__HIP_DEVICE_COMPILE__) && __has_builtin(__builtin_amdgcn_wmma_f32_16x16x4_f32)
#define USE_WMMA_RED 1
#else
#define USE_WMMA_RED 0
#endif

__device__ __forceinline__ float softplusf(float x) {
  // jax.nn.softplus(x) = max(x,0) + log1p(exp(-|x|))  (numerically stable)
  return fmaxf(x, 0.0f) + log1pf(expf(-fabsf(x)));
}
__device__ __forceinline__ float sigmoidf_(float x) {
  return 1.0f / (1.0f + expf(-x));
}
// wave32 butterfly reduction: every lane ends with the wave sum (deterministic)
__device__ __forceinline__ float waveSum(float v) {
  v += __shfl_xor(v, 16, 32);
  v += __shfl_xor(v,  8, 32);
  v += __shfl_xor(v,  4, 32);
  v += __shfl_xor(v,  2, 32);
  v += __shfl_xor(v,  1, 32);
  return v;
}

// ---------------------------------------------------------------------------
__global__ void yolo_zero(unsigned* __restrict__ occ) {
  int i = blockIdx.x * blockDim.x + threadIdx.x;
  if (i < OCCW) occ[i] = 0u;
}

// ---------------------------------------------------------------------------
// Per-target work: occupancy bit, obj / coord / class partial losses.
__global__ void yolo_targets(const float* __restrict__ pred,
                             const float* __restrict__ tgt,
                             unsigned*    __restrict__ occ,
                             float*       __restrict__ tpart) {
  __shared__ float lds[8];
  const int t = blockIdx.x * blockDim.x + threadIdx.x;

  float obj = 0.0f, coord = 0.0f, cls = 0.0f;
  if (t < NT) {
    const float* tr = tgt + (size_t)t * 5;
    const int   b   = t / T;
    const int   cid = (int)tr[0];
    const float xg  = tr[1] * (float)G;
    const float yg  = tr[2] * (float)G;
    const float wg  = tr[3] * (float)G;
    const float hg  = tr[4] * (float)G;
    const int gx = min(max((int)xg, 0), G - 1);
    const int gy = min(max((int)yg, 0), G - 1);

    const size_t cell = ((size_t)b * G + gy) * G + gx;
    atomicOr(&occ[cell >> 5], 1u << (int)(cell & 31));   // exact OR: deterministic

    const float* p = pred + cell * (size_t)C5;
    obj = softplusf(-p[0]);

    const float dx = sigmoidf_(p[1]) - (xg - (float)gx);
    const float dy = sigmoidf_(p[2]) - (yg - (float)gy);
    const float dw = p[3] - wg;
    const float dh = p[4] - hg;
    coord = dx * dx + dy * dy + dw * dw + dh * dh;

    float cs = 0.0f;
#pragma unroll 4
    for (int j = 0; j < NC; ++j) cs += softplusf(p[5 + j]);
    if (cid >= 0 && cid < NC) cs -= p[5 + cid];
    cls = cs;
  }

  // deterministic block reduction of 3 scalars
  float v3[3] = {obj, coord, cls};
  for (int k = 0; k < 3; ++k) {
    float v = waveSum(v3[k]);
    if ((threadIdx.x & 31) == 0) lds[threadIdx.x >> 5] = v;
    __syncthreads();
    if (threadIdx.x == 0) {
      float s = 0.0f;
      for (int w = 0; w < (int)(blockDim.x >> 5); ++w) s += lds[w];
      tpart[blockIdx.x * 3 + k] = s;
    }
    __syncthreads();
  }
}

// ---------------------------------------------------------------------------
// Streaming noobj pass.  Each block owns a contiguous 4096-cell chunk;
// every thread handles exactly PER cells -> no divergence, EXEC all-ones,
// satisfying the WMMA execution requirement.
__global__ void yolo_noobj(const float*    __restrict__ pred,
                           const unsigned* __restrict__ occ,
                           float*          __restrict__ npart) {
  __shared__ float lds[TPB / 32];
  const int base = blockIdx.x * (TPB * PER);

#if USE_WMMA_RED
  v8f acc = {0.f, 0.f, 0.f, 0.f, 0.f, 0.f, 0.f, 0.f};
  const v2f ones = {1.0f, 1.0f};
  const float scale = 1.0f / 16.0f;     // sum(acc) == 16 * true sum
#else
  float accs = 0.0f;
  const float scale = 1.0f;
#endif
  float pair0 = 0.0f;
  (void)pair0;

#pragma unroll
  for (int i = 0; i < PER; ++i) {
    const int cell = base + i * TPB + (int)threadIdx.x;
    const float p0 = pred[(size_t)cell * C5];
    // speculative prefetch of next tile (global_prefetch_b8); OOB is dropped
    __builtin_prefetch(&pred[(size_t)(cell + TPB) * C5], 0, 1);
    const unsigned wbits = occ[cell >> 5];
    const float v = ((wbits >> (cell & 31)) & 1u) ? 0.0f : softplusf(p0);
#if USE_WMMA_RED
    if ((i & 1) == 0) {
      pair0 = v;
    } else {
      v2f bv = {pair0, v};
      // D = ones(16x4) x B(4x16) + C : matrix-pipe accumulation, co-executes
      // with the softplus transcendentals on the VALU.
      acc = __builtin_amdgcn_wmma_f32_16x16x4_f32(
          /*neg_a=*/false, ones, /*neg_b=*/false, bv,
          /*c_mod=*/(short)0, acc, /*reuse_a=*/false, /*reuse_b=*/false);
    }
#else
    accs += v;
#endif
  }

#if USE_WMMA_RED
  float s = acc[0] + acc[1] + acc[2] + acc[3] + acc[4] + acc[5] + acc[6] + acc[7];
#else
  float s = accs;
#endif
  s = waveSum(s) * scale;

  if ((threadIdx.x & 31) == 0) lds[threadIdx.x >> 5] = s;
  __syncthreads();
  if (threadIdx.x == 0) {
    float r = 0.0f;
    for (int w = 0; w < TPB / 32; ++w) r += lds[w];
    npart[blockIdx.x] = r;
  }
}

// ---------------------------------------------------------------------------
// Fixed-order serial combine: bit-for-bit deterministic across replays.
__global__ void yolo_final(const float* __restrict__ tpart, int ntb,
                           const float* __restrict__ npart, int nnb,
                           float* __restrict__ out) {
  if (blockIdx.x == 0 && threadIdx.x == 0) {
    float obj = 0.f, coord = 0.f, cls = 0.f, noobj = 0.f;
    for (int i = 0; i < ntb; ++i) {
      obj   += tpart[3 * i + 0];
      coord += tpart[3 * i + 1];
      cls   += tpart[3 * i + 2];
    }
    for (int i = 0; i < nnb; ++i) noobj += npart[i];
    const float invB = 1.0f / (float)B;
    obj   *= invB;
    noobj  = LAMBDA_NOOBJ * noobj * invB;
    coord  = LAMBDA_COORD * coord * invB;
    cls   *= invB;
    out[0] = obj + noobj + coord + cls;
    out[1] = obj;
    out[2] = noobj;
    out[3] = coord;
    out[4] = cls;
  }
}

// ---------------------------------------------------------------------------
extern "C" void kernel_launch(void* const* d_in, const int* in_sizes, int n_in,
                              void* d_out, int out_size, void* d_ws, size_t ws_size,
                              hipStream_t stream) {
  (void)in_sizes; (void)n_in; (void)out_size; (void)ws_size;
  const float* pred = (const float*)d_in[0];
  const float* tgt  = (const float*)d_in[1];
  float* out = (float*)d_out;

  // workspace layout: [occ mask: 12800 u32][tpart: 64 f32][npart: 100 f32]
  unsigned* occ   = (unsigned*)d_ws;
  float*    tpart = (float*)((char*)d_ws + (size_t)OCCW * sizeof(unsigned));
  float*    npart = tpart + 64;

  const int ntb = (NT + 255) / 256;   // 13 target blocks

  yolo_zero   <<<(OCCW + 255) / 256, 256, 0, stream>>>(occ);
  yolo_targets<<<ntb,  256, 0, stream>>>(pred, tgt, occ, tpart);
  yolo_noobj  <<<NBLK, TPB, 0, stream>>>(pred, occ, npart);
  yolo_final  <<<1, 32, 0, stream>>>(tpart, ntb, npart, NBLK, out);
}